// SSIM_66013647339611
// MI455X (gfx1250) — compile-verified
//
#include <hip/hip_runtime.h>

// ---------------------------------------------------------------------------
// Fused SSIM on gfx1250: separable 11-tap Gaussian conv as banded matmuls on
// the f32 WMMA pipe (V_WMMA_F32_16X16X4_F32), full fp32 precision.
// ---------------------------------------------------------------------------

typedef __attribute__((ext_vector_type(2))) float v2f;
typedef __attribute__((ext_vector_type(8))) float v8f;

#define WIN      11
#define PADW     5
#define RSTRIP   128                 // output rows per workgroup
#define SROWS    144                 // staged rows = RSTRIP + 16 (halo + pad)
#define XCOLS    28                  // staged cols = 16 + 2*5 halo + 2 pad (K mult of 4)
#define TSTR     20                  // padded T row stride (floats): bank-conflict free
#define NTHREADS 256                 // 8 wave32 waves
#define SSIM_C1  1.0e-4f
#define SSIM_C2  9.0e-4f
#define IMG      512
#define NCIMG    (32 * 3)
#define COLSTRIPS (IMG / 16)         // 32
#define ROWSTRIPS (IMG / RSTRIP)     // 4
#define NBLOCKS  (NCIMG * COLSTRIPS * ROWSTRIPS)  // 12288 partials (48 KB in d_ws)

__device__ __forceinline__ v8f wmma4(v2f a, v2f b, v8f c) {
  // D = A(16x4,f32) * B(4x16,f32) + C(16x16,f32)
  return __builtin_amdgcn_wmma_f32_16x16x4_f32(
      /*neg_a=*/false, a, /*neg_b=*/false, b,
      /*c_mod=*/(short)0, c, /*reuse_a=*/false, /*reuse_b=*/false);
}

__global__ __launch_bounds__(NTHREADS)
void ssim_fused(const float* __restrict__ img1, const float* __restrict__ img2,
                float* __restrict__ partials) {
  __shared__ float xs[2][SROWS * XCOLS];   // staged img1/img2 strip (32.3 KB)
  __shared__ float ts[5][SROWS * TSTR];    // horizontal-pass intermediates (57.6 KB)
  __shared__ float red[NTHREADS];

  const int tid  = threadIdx.x;
  const int wave = tid >> 5;               // wave32
  const int lane = tid & 31;
  const int n16  = lane & 15;              // M (A rows) / N (B cols) index
  const int kb   = (lane >> 4) << 1;       // K base of this lane half: 0 or 2

  const int bid = blockIdx.x;
  const int nc  = bid / (COLSTRIPS * ROWSTRIPS);
  const int s   = bid % (COLSTRIPS * ROWSTRIPS);
  const int rs  = (s / COLSTRIPS) * RSTRIP;
  const int cs  = (s % COLSTRIPS) * 16;
  const size_t base = (size_t)nc * IMG * IMG;

  // ---- stage halo'd strips (zero outside image == reference zero padding) ----
  for (int i = tid; i < 2 * SROWS * XCOLS; i += NTHREADS) {
    const int im = i / (SROWS * XCOLS);
    const int r  = (i / XCOLS) % SROWS;
    const int c  = i % XCOLS;
    const int gr = rs - (PADW + 1) + r;    // strip row 0 == image row rs-6
    const int gc = cs - (PADW + 1) + c;    // strip col 0 == image col cs-6
    float v = 0.f;
    if ((unsigned)gr < IMG && (unsigned)gc < IMG) {
      const float* src = im ? img2 : img1;
      v = src[base + (size_t)gr * IMG + gc];
    }
    xs[im][r * XCOLS + c] = v;
  }

  // ---- per-lane banded Gaussian fragments (serve as B in pass 1, A in pass 2)
  float z = 0.f;
#pragma unroll
  for (int t = 0; t < WIN; ++t) { float d = (float)(t - PADW); z += expf(-d * d * (1.f / 4.5f)); }
  const float ginv = 1.f / z;
  v2f gf[7];
#pragma unroll
  for (int kc = 0; kc < 7; ++kc) {
#pragma unroll
    for (int j = 0; j < 2; ++j) {
      const int i = 4 * kc + kb + j - n16 - 1;   // band: g[k - idx - 1]
      float g = 0.f;
      if (i >= 0 && i < WIN) { float d = (float)(i - PADW); g = expf(-d * d * (1.f / 4.5f)) * ginv; }
      gf[kc][j] = g;
    }
  }

  __syncthreads();

  // ---- pass 1 (horizontal): T[mb*16+m, n] = sum_k X[row, k] * g[k-n-1] ----
  // A = data (ds_load_b64 fills K-even/K-odd VGPR pair), B = Gaussian fragments.
  for (int mb = wave; mb < SROWS / 16; mb += NTHREADS / 32) {
    const int mrow = mb * 16 + n16;
    v8f aMu1 = {0,0,0,0,0,0,0,0}, aMu2 = {0,0,0,0,0,0,0,0};
    v8f aXX  = {0,0,0,0,0,0,0,0}, aYY  = {0,0,0,0,0,0,0,0}, aXY = {0,0,0,0,0,0,0,0};
#pragma unroll
    for (int kc = 0; kc < 7; ++kc) {
      const int k0 = 4 * kc + kb;                 // even -> 8B aligned
      const v2f ax = *(const v2f*)&xs[0][mrow * XCOLS + k0];
      const v2f ay = *(const v2f*)&xs[1][mrow * XCOLS + k0];
      const v2f axx = ax * ax, ayy = ay * ay, axy = ax * ay;
      aMu1 = wmma4(ax,  gf[kc], aMu1);
      aMu2 = wmma4(ay,  gf[kc], aMu2);
      aXX  = wmma4(axx, gf[kc], aXX);
      aYY  = wmma4(ayy, gf[kc], aYY);
      aXY  = wmma4(axy, gf[kc], aXY);
    }
    const int rlo = mb * 16 + ((lane >> 4) << 3);  // D rows: lanes<16 -> M=j, else M=8+j
#pragma unroll
    for (int j = 0; j < 8; ++j) {
      const int row = rlo + j;
      ts[0][row * TSTR + n16] = aMu1[j];
      ts[1][row * TSTR + n16] = aMu2[j];
      ts[2][row * TSTR + n16] = aXX[j];
      ts[3][row * TSTR + n16] = aYY[j];
      ts[4][row * TSTR + n16] = aXY[j];
    }
  }
  __syncthreads();

  // ---- pass 2 (vertical): Y[m,n] = sum_k g[k-m-1] * T[q+k, n] ----
  // A = Gaussian fragments (same gf!), B = T rows from LDS.
  const int q = wave * 16;   // this wave's 16-row output tile uses strip rows q..q+27
  v8f m1 = {0,0,0,0,0,0,0,0}, m2 = {0,0,0,0,0,0,0,0};
  v8f sxx = {0,0,0,0,0,0,0,0}, syy = {0,0,0,0,0,0,0,0}, sxy = {0,0,0,0,0,0,0,0};
#pragma unroll
  for (int kc = 0; kc < 7; ++kc) {
    const int k0 = q + 4 * kc + kb;
    v2f b;
    b.x = ts[0][(k0) * TSTR + n16]; b.y = ts[0][(k0 + 1) * TSTR + n16];
    m1 = wmma4(gf[kc], b, m1);
    b.x = ts[1][(k0) * TSTR + n16]; b.y = ts[1][(k0 + 1) * TSTR + n16];
    m2 = wmma4(gf[kc], b, m2);
    b.x = ts[2][(k0) * TSTR + n16]; b.y = ts[2][(k0 + 1) * TSTR + n16];
    sxx = wmma4(gf[kc], b, sxx);
    b.x = ts[3][(k0) * TSTR + n16]; b.y = ts[3][(k0 + 1) * TSTR + n16];
    syy = wmma4(gf[kc], b, syy);
    b.x = ts[4][(k0) * TSTR + n16]; b.y = ts[4][(k0 + 1) * TSTR + n16];
    sxy = wmma4(gf[kc], b, sxy);
  }

  // ---- elementwise SSIM map + in-block reduction ----
  float lsum = 0.f;
#pragma unroll
  for (int j = 0; j < 8; ++j) {
    const float mu1 = m1[j], mu2 = m2[j];
    const float mu1sq = mu1 * mu1, mu2sq = mu2 * mu2, mu12 = mu1 * mu2;
    const float s1 = sxx[j] - mu1sq;
    const float s2 = syy[j] - mu2sq;
    const float s12 = sxy[j] - mu12;
    lsum += ((2.f * mu12 + SSIM_C1) * (2.f * s12 + SSIM_C2)) /
            ((mu1sq + mu2sq + SSIM_C1) * (s1 + s2 + SSIM_C2));
  }
  red[tid] = lsum;
  __syncthreads();
  if (tid == 0) {
    float t = 0.f;
    for (int i = 0; i < NTHREADS; ++i) t += red[i];
    partials[bid] = t;
  }
}

// Deterministic fixed-order final reduction of per-WG partials.
__global__ void ssim_reduce(const float* __restrict__ partials, float* __restrict__ out,
                            int n, double invN) {
  __shared__ double sh[256];
  double acc = 0.0;
  for (int i = threadIdx.x; i < n; i += 256) acc += (double)partials[i];
  sh[threadIdx.x] = acc;
  __syncthreads();
  if (threadIdx.x == 0) {
    double t = 0.0;
    for (int i = 0; i < 256; ++i) t += sh[i];
    out[0] = (float)(t * invN);
  }
}

extern "C" void kernel_launch(void* const* d_in, const int* in_sizes, int n_in,
                              void* d_out, int out_size, void* d_ws, size_t ws_size,
                              hipStream_t stream) {
  (void)in_sizes; (void)n_in; (void)out_size; (void)ws_size;
  const float* img1 = (const float*)d_in[0];
  const float* img2 = (const float*)d_in[1];
  float* out = (float*)d_out;
  float* partials = (float*)d_ws;   // NBLOCKS floats = 48 KB scratch

  ssim_fused<<<NBLOCKS, NTHREADS, 0, stream>>>(img1, img2, partials);
  const double invN = 1.0 / ((double)NCIMG * IMG * IMG);
  ssim_reduce<<<1, 256, 0, stream>>>(partials, out, NBLOCKS, invN);
}